// AttentionPooling_26233660244214
// MI455X (gfx1250) — compile-verified
//
#include <hip/hip_runtime.h>
#include <math.h>

typedef __attribute__((ext_vector_type(2))) float v2f;
typedef __attribute__((ext_vector_type(4))) float v4f;
typedef __attribute__((ext_vector_type(8))) float v8f;

#define NODE_DIM 256
#define LDS_STRIDE 264      // padded row stride (floats) to spread LDS banks
#define TILES_PER_WAVE 8

__device__ __forceinline__ float readlane_f32(float v, int l) {
    return __int_as_float(__builtin_amdgcn_readlane(__float_as_int(v), l));
}

__global__ __launch_bounds__(256) void zero_out_kernel(float* __restrict__ out, int n4) {
    int i = blockIdx.x * blockDim.x + threadIdx.x;
    if (i < n4) {
        v4f z = {};
        *(v4f*)(out + 4 * (size_t)i) = z;
    }
}

__global__ __launch_bounds__(32) void attn_pool_kernel(
    const float* __restrict__ feats,
    const int*   __restrict__ batch_idx,
    const float* __restrict__ Wa,
    const float* __restrict__ ba,
    const float* __restrict__ Wm,
    const float* __restrict__ bmask,
    float* __restrict__ out,
    int n_nodes, int n_tiles)
{
    __shared__ float smem[16 * LDS_STRIDE];   // 16.5 KB, wave-private (1 wave/WG)

    const int lane = threadIdx.x;             // 0..31 (wave32)
    const int row  = lane & 15;               // A-matrix row (node within tile)
    const int koff = (lane >> 4) * 2;         // 0 for lanes 0-15, 2 for lanes 16-31
    const int sel  = row;                     // B column this lane would feed

    const float bias_a = ba[0];
    const float bias_m = bmask[0];

    const int tile0    = blockIdx.x * TILES_PER_WAVE;
    const int tile_end = (tile0 + TILES_PER_WAVE < n_tiles) ? tile0 + TILES_PER_WAVE : n_tiles;

    float acc[8];
    #pragma unroll
    for (int j = 0; j < 8; ++j) acc[j] = 0.f;
    int cur = -1;                             // current segment held in acc

    for (int t = tile0; t < tile_end; ++t) {
        const int base = t * 16;

        // Clamped node index for this lane's row; tail rows get fin=0 below.
        int nidx = base + row;
        if (nidx > n_nodes - 1) nidx = n_nodes - 1;
        const int bval = batch_idx[nidx];

        const float* rowp = feats + (size_t)nidx * NODE_DIM + koff;
        float*       srow = &smem[row * LDS_STRIDE + koff];
        const float* wp   = (sel == 1) ? Wm : Wa;
        const v2f    zero2 = {};

        // ---- Phase 1: scores[16,2] = X_tile[16,256] @ [Wa|Wm] via V_WMMA_F32_16X16X4_F32
        v8f c0 = {}; v8f c1 = {};
        #pragma unroll 4
        for (int k = 0; k < 64; k += 2) {
            // A operand (16x4 f32): lane holds float2 of its row at cols 4k+koff
            v2f a0 = __builtin_nontemporal_load((const v2f*)(rowp + 4 * k));
            v2f a1 = __builtin_nontemporal_load((const v2f*)(rowp + 4 * k + 4));
            // stage into LDS for phase 2 (column-owned re-read)
            *(v2f*)(srow + 4 * k)     = a0;
            *(v2f*)(srow + 4 * k + 4) = a1;
            // B operand (4x16 f32): col 0 = W_attn chunk, col 1 = W_mask chunk, rest 0
            v2f b0 = *(const v2f*)(wp + 4 * k + koff);
            v2f b1 = *(const v2f*)(wp + 4 * k + 4 + koff);
            if (sel > 1) { b0 = zero2; b1 = zero2; }
            c0 = __builtin_amdgcn_wmma_f32_16x16x4_f32(false, a0, false, b0, (short)0, c0, false, false);
            c1 = __builtin_amdgcn_wmma_f32_16x16x4_f32(false, a1, false, b1, (short)0, c1, false, false);
        }
        v8f c = c0 + c1;

        // ---- Extract per-node final score: (attn+b) * sigmoid(mask+b)
        // C layout: VGPR r -> lanes 0-15: M=r,N=lane ; lanes 16-31: M=r+8,N=lane-16
        // Constant source lanes -> v_readlane_b32 (no ds_bpermute traffic).
        float fin[16];
        #pragma unroll
        for (int r = 0; r < 8; ++r) {
            float alo = readlane_f32(c[r], 0);
            float mlo = readlane_f32(c[r], 1);
            float ahi = readlane_f32(c[r], 16);
            float mhi = readlane_f32(c[r], 17);
            float flo = (alo + bias_a) * (1.f / (1.f + expf(-(mlo + bias_m))));
            float fhi = (ahi + bias_a) * (1.f / (1.f + expf(-(mhi + bias_m))));
            fin[r]     = (base + r     < n_nodes) ? flo : 0.f;
            fin[r + 8] = (base + r + 8 < n_nodes) ? fhi : 0.f;
        }

        // ---- Phase 2: weighted accumulate into per-wave segment accumulator.
        // Lane owns dims [lane*8, lane*8+8). batch_idx sorted -> flush on change.
        #pragma unroll
        for (int m = 0; m < 16; ++m) {
            int bm = __builtin_amdgcn_readlane(bval, m);   // uniform (SGPR) broadcast
            if (bm != cur) {
                if (cur >= 0) {
                    float* dst = out + (size_t)cur * NODE_DIM + lane * 8;
                    #pragma unroll
                    for (int j = 0; j < 8; ++j) unsafeAtomicAdd(dst + j, acc[j]);
                }
                cur = bm;
                #pragma unroll
                for (int j = 0; j < 8; ++j) acc[j] = 0.f;
            }
            const float* mrow = &smem[m * LDS_STRIDE + lane * 8];
            v4f x0 = *(const v4f*)(mrow);
            v4f x1 = *(const v4f*)(mrow + 4);
            float fm = fin[m];
            acc[0] += fm * x0.x; acc[1] += fm * x0.y;
            acc[2] += fm * x0.z; acc[3] += fm * x0.w;
            acc[4] += fm * x1.x; acc[5] += fm * x1.y;
            acc[6] += fm * x1.z; acc[7] += fm * x1.w;
        }
    }

    if (cur >= 0) {
        float* dst = out + (size_t)cur * NODE_DIM + lane * 8;
        #pragma unroll
        for (int j = 0; j < 8; ++j) unsafeAtomicAdd(dst + j, acc[j]);
    }
}

extern "C" void kernel_launch(void* const* d_in, const int* in_sizes, int n_in,
                              void* d_out, int out_size, void* d_ws, size_t ws_size,
                              hipStream_t stream) {
    const float* feats     = (const float*)d_in[0];
    const int*   batch_idx = (const int*)d_in[1];
    const float* Wa        = (const float*)d_in[2];
    const float* ba        = (const float*)d_in[3];
    const float* Wm        = (const float*)d_in[4];
    const float* bm        = (const float*)d_in[5];
    float*       out       = (float*)d_out;

    const int n_nodes = in_sizes[1];           // batch_idx element count

    // Zero the (poisoned) output before atomic accumulation.
    int n4 = out_size / 4;
    zero_out_kernel<<<(n4 + 255) / 256, 256, 0, stream>>>(out, n4);

    const int n_tiles = (n_nodes + 15) / 16;
    const int n_waves = (n_tiles + TILES_PER_WAVE - 1) / TILES_PER_WAVE;
    attn_pool_kernel<<<n_waves, 32, 0, stream>>>(feats, batch_idx, Wa, ba, Wm, bm,
                                                 out, n_nodes, n_tiles);
}